// IRBlock_45870250721473
// MI455X (gfx1250) — compile-verified
//
#include <hip/hip_runtime.h>

// CDNA5 / gfx1250 implicit-GEMM 3x3 conv block using V_WMMA_F32_16X16X4_F32.
// Two passes: y1 = PReLU1(BN1(conv1(BN0(x))));  out = PReLU2(BN2(conv2(y1)) + x).
//
// LDS layout: input tile FIRST (all hot-loop B displacements fit the 16-bit DS
// offset field -> no per-load address VALU), weights second. Input tile is
// channel-pair interleaved ([c/2][ih][iw][2]) so each WMMA B fragment is one
// aligned ds_load_b64. Each block processes 4 batch images with ONE weight
// stage (4x less L2 weight traffic); weights are read with global_load_b128.

typedef __attribute__((ext_vector_type(2))) float v2f;
typedef __attribute__((ext_vector_type(8))) float v8f;

#define C_CH 64
#define HGT 112
#define WID 112
#define TH 8          // output rows per block
#define TW 16         // output cols per block
#define NPB 4         // batch images per block (weight-stage amortization)
#define IHALO 10      // TH + 2
#define IWALO 18      // TW + 2
#define W_LDS_FLOATS (9 * 64 * 64)          // 36864 floats, layout [tap][o][c]
#define I_LDS_FLOATS (C_CH * IHALO * IWALO) // 11520 floats, layout [c/2][ih][iw][2]
#define PAIR_STRIDE (IHALO * IWALO * 2)     // floats per channel pair = 360
#define LDS_BYTES ((W_LDS_FLOATS + I_LDS_FLOATS) * 4) // 193536 B < 320KB/WGP

__global__ __launch_bounds__(64)
void bn_prep_kernel(const float* g0, const float* b0, const float* m0, const float* v0,
                    const float* g1, const float* b1, const float* m1, const float* v1,
                    const float* g2, const float* b2, const float* m2, const float* v2,
                    float* pbuf) {
    int i = threadIdx.x;
    if (i < C_CH) {
        float s0 = g0[i] * rsqrtf(v0[i] + 1e-5f);
        pbuf[0 * 64 + i] = s0;
        pbuf[1 * 64 + i] = b0[i] - m0[i] * s0;
        float s1 = g1[i] * rsqrtf(v1[i] + 1e-5f);
        pbuf[2 * 64 + i] = s1;
        pbuf[3 * 64 + i] = b1[i] - m1[i] * s1;
        float s2 = g2[i] * rsqrtf(v2[i] + 1e-5f);
        pbuf[4 * 64 + i] = s2;
        pbuf[5 * 64 + i] = b2[i] - m2[i] * s2;
    }
}

// PRE:  apply per-input-channel affine (BN) while staging input into LDS.
// RES:  add residual tensor before the final PReLU.
template <bool PRE, bool RES>
__global__ __launch_bounds__(256)
void conv3x3_wmma_kernel(const float* __restrict__ in, const float* __restrict__ wgt,
                         const float* __restrict__ pre_s, const float* __restrict__ pre_b,
                         const float* __restrict__ post_s, const float* __restrict__ post_b,
                         const float* __restrict__ prelu_a, const float* __restrict__ res,
                         float* __restrict__ out) {
    extern __shared__ float smem[];
    float* ilds = smem;                   // [32 cpair][10 ih][18 iw][2 c&1]  @ 0
    float* wlds = smem + I_LDS_FLOATS;    // [9 tap][64 o][64 c]              @ 46080 B

    const int tid = threadIdx.x;
    const int w0 = blockIdx.x * TW;
    const int h0 = blockIdx.y * TH;

    // --- stage weights once per block: wlds[(r*64+o)*64+c] = wgt[o*576+c*9+r]
    // 128-bit coalesced global reads, scattered LDS writes
    for (int t4 = tid; t4 < W_LDS_FLOATS / 4; t4 += 256) {
        const float4 wv = ((const float4*)wgt)[t4];
        const float vals[4] = {wv.x, wv.y, wv.z, wv.w};
        int base = t4 * 4;
        int o = base / 576;
        int rem = base - o * 576;
        int c = rem / 9;
        int r = rem - c * 9;
#pragma unroll
        for (int k = 0; k < 4; ++k) {
            wlds[(r * 64 + o) * 64 + c] = vals[k];
            if (++r == 9) { r = 0; if (++c == 64) { c = 0; ++o; } }
        }
    }

    // wave -> (c_out group, row group); lane -> (K-half, N column)
    const int lane = tid & 31;
    const int wave = tid >> 5;
    const int cog  = wave & 3;   // c_out group of 16
    const int rgrp = wave >> 2;  // row group of 4
    const int half = lane >> 4;  // K pair select: channels 4s+{0,1} vs 4s+{2,3}
    const int lm   = lane & 15;  // A: M row ; B/D: N column
    const int rbase = rgrp * 4;
    const v8f zero8 = {0.f, 0.f, 0.f, 0.f, 0.f, 0.f, 0.f, 0.f};

    for (int ni = 0; ni < NPB; ++ni) {
        const int n = blockIdx.z * NPB + ni;

        // --- stage input halo with pre-BN folded; zeros in the padding region
        const float* inN = in + (size_t)n * C_CH * HGT * WID;
        for (int t = tid; t < I_LDS_FLOATS; t += 256) {
            int c = t / (IHALO * IWALO);
            int rem = t - c * (IHALO * IWALO);
            int ih = rem / IWALO;
            int iw = rem - ih * IWALO;
            int gh = h0 - 1 + ih;
            int gw = w0 - 1 + iw;
            float v = 0.0f;
            if ((unsigned)gh < (unsigned)HGT && (unsigned)gw < (unsigned)WID) {
                v = inN[(c * HGT + gh) * WID + gw];
                if (PRE) v = v * pre_s[c] + pre_b[c];
            }
            // channel-pair interleaved destination
            ilds[(((c >> 1) * IHALO + ih) * IWALO + iw) * 2 + (c & 1)] = v;
        }
        __syncthreads();   // staging (weights on first pass + input) visible

        v8f acc[4];
#pragma unroll
        for (int j = 0; j < 4; ++j) acc[j] = zero8;

        // 9 taps x 16 K-steps x 4 row-tiles = 576 v_wmma_f32_16x16x4_f32/wave
        for (int r = 0; r < 9; ++r) {
            const int kh = r / 3;
            const int kw = r - kh * 3;
            // A: wlds[tap r][o = cog*16+lm][c = 4s + 2*half .. +1]  (b64)
            const float* wrow = &wlds[(r * 64 + cog * 16 + lm) * 64 + 2 * half];
            // B: channel pair cp = 2s+half, pixel (rbase+j+kh, lm+kw)  (b64)
            const float* ibase =
                &ilds[((half * IHALO + rbase + kh) * IWALO + lm + kw) * 2];
#pragma unroll 4
            for (int s = 0; s < 16; ++s) {
                v2f a = *(const v2f*)(wrow + s * 4);    // ds_load_b64, reused 4x
                const float* ip = ibase + s * 2 * PAIR_STRIDE;
#pragma unroll
                for (int j = 0; j < 4; ++j) {
                    v2f b = *(const v2f*)(ip + j * (IWALO * 2));  // ds_load_b64
                    acc[j] = __builtin_amdgcn_wmma_f32_16x16x4_f32(
                        false, a, false, b, (short)0, acc[j], false, false);
                }
            }
        }

        // --- epilogue: BN affine (+residual) + PReLU, scatter to NCHW ---
        // D layout: lanes 0-15 M=e, lanes 16-31 M=e+8, N=lm
#pragma unroll
        for (int j = 0; j < 4; ++j) {
            const int gh = h0 + rbase + j;
#pragma unroll
            for (int e = 0; e < 8; ++e) {
                const int o = cog * 16 + e + 8 * half;
                const size_t idx =
                    ((size_t)(n * C_CH + o) * HGT + gh) * WID + (w0 + lm);
                float v = acc[j][e] * post_s[o] + post_b[o];
                if (RES) v += res[idx];
                const float al = prelu_a[o];
                v = (v >= 0.0f) ? v : al * v;
                out[idx] = v;
            }
        }
        __syncthreads();   // all K-loop reads done before next input stage
    }
}

extern "C" void kernel_launch(void* const* d_in, const int* in_sizes, int n_in,
                              void* d_out, int out_size, void* d_ws, size_t ws_size,
                              hipStream_t stream) {
    (void)in_sizes; (void)n_in; (void)out_size; (void)ws_size;

    const float* x  = (const float*)d_in[0];
    const float* w1 = (const float*)d_in[1];
    const float* w2 = (const float*)d_in[2];
    const float* bn0_g = (const float*)d_in[3];
    const float* bn0_b = (const float*)d_in[4];
    const float* bn0_m = (const float*)d_in[5];
    const float* bn0_v = (const float*)d_in[6];
    const float* bn1_g = (const float*)d_in[7];
    const float* bn1_b = (const float*)d_in[8];
    const float* bn1_m = (const float*)d_in[9];
    const float* bn1_v = (const float*)d_in[10];
    const float* bn2_g = (const float*)d_in[11];
    const float* bn2_b = (const float*)d_in[12];
    const float* bn2_m = (const float*)d_in[13];
    const float* bn2_v = (const float*)d_in[14];
    const float* prelu1_a = (const float*)d_in[15];
    const float* prelu2_a = (const float*)d_in[16];

    float* pbuf = (float*)d_ws;                       // 6*64 floats of BN affines
    float* y1   = (float*)((char*)d_ws + 2048);       // intermediate activation

    bn_prep_kernel<<<1, 64, 0, stream>>>(bn0_g, bn0_b, bn0_m, bn0_v,
                                         bn1_g, bn1_b, bn1_m, bn1_v,
                                         bn2_g, bn2_b, bn2_m, bn2_v, pbuf);

    dim3 grid(WID / TW, HGT / TH, 64 / NPB);   // 7 x 14 x 16 blocks
    dim3 block(256);

    // pass 1: y1 = PReLU1(BN1(conv1(BN0(x))))
    conv3x3_wmma_kernel<true, false><<<grid, block, LDS_BYTES, stream>>>(
        x, w1, pbuf + 0, pbuf + 64, pbuf + 128, pbuf + 192, prelu1_a, nullptr, y1);

    // pass 2: out = PReLU2(BN2(conv2(y1)) + x)
    conv3x3_wmma_kernel<false, true><<<grid, block, LDS_BYTES, stream>>>(
        y1, w2, nullptr, nullptr, pbuf + 256, pbuf + 320, prelu2_a, x, (float*)d_out);
}